// APG_Linear_32530082299935
// MI455X (gfx1250) — compile-verified
//
#include <hip/hip_runtime.h>

// Problem constants (match reference)
#define NB       131072
#define IN_FEAT  128
#define OUT_FEAT 128
#define RANK     32

typedef __attribute__((ext_vector_type(2))) float v2f;
typedef __attribute__((ext_vector_type(8))) float v8f;

// LDS strides in dwords (padded for bank-conflict-free wave32 access patterns)
#define UT_STRIDE 130   // sUt[32][130]  : sUt[n][k] = U[k][n]
#define VT_STRIDE 34    // sVt[128][34]  : sVt[o][r] = V[r][o]
#define H_STRIDE  34    // per-wave h/h2 tile [16][34]

__global__ __launch_bounds__(256) void apg_linear_kernel(
    const float* __restrict__ inp,   // [B,128]
    const float* __restrict__ gw,    // [B,1024] per-sample 32x32 weights
    const float* __restrict__ U,     // [128,32]
    const float* __restrict__ V,     // [32,128]
    const float* __restrict__ bias,  // [128]
    float* __restrict__ out)         // [B,128]
{
    __shared__ float sUt[RANK * UT_STRIDE];      // 16.6 KB
    __shared__ float sVt[OUT_FEAT * VT_STRIDE];  // 17.4 KB
    __shared__ float sBias[OUT_FEAT];            // 0.5 KB
    __shared__ float sH[8][16 * H_STRIDE];       // 17.4 KB (per-wave h, reused for h2)

    const int tid  = threadIdx.x;
    const int wave = tid >> 5;
    const int lane = tid & 31;

    // ---- Stage U^T, V^T, bias into LDS (tiny, cache-resident; regular loads) ----
    for (int idx = tid; idx < IN_FEAT * RANK; idx += 256) {
        int k = idx >> 5, n = idx & 31;          // U is [k][n]
        sUt[n * UT_STRIDE + k] = U[idx];
    }
    for (int idx = tid; idx < RANK * OUT_FEAT; idx += 256) {
        int r = idx >> 7, o = idx & 127;         // V is [r][o]
        sVt[o * VT_STRIDE + r] = V[idx];
    }
    if (tid < OUT_FEAT) sBias[tid] = bias[tid];
    __syncthreads();

    const int tile = blockIdx.x * 8 + wave;      // one 16-sample tile per wave
    const int row0 = tile * 16;

    const int fm = lane & 15;                    // A-frag row / B-frag column
    const int kh = lane >> 4;                    // K-half selector (f32 16x16x4 layout)

    // =====================================================================
    // Step 1: h[16x32] = inp_tile[16x128] @ U[128x32]
    //   V_WMMA_F32_16X16X4_F32: 32 K-steps x 2 N-tiles.
    //   A frag (16x4 f32): lane(fm,kh) holds {inp[fm][4k+2kh], inp[fm][4k+2kh+1]}
    //   B frag (4x16 f32): lane(fm,kh) holds {U[4k+2kh][col], U[4k+2kh+1][col]}
    // =====================================================================
    v8f c0 = {};  // columns 0..15
    v8f c1 = {};  // columns 16..31
    const float* inpRow = inp + (size_t)(row0 + fm) * IN_FEAT;
    #pragma unroll 4
    for (int k = 0; k < IN_FEAT / 4; ++k) {
        const int kk = 4 * k + 2 * kh;
        // inp is streamed exactly once -> non-temporal
        v2f a = __builtin_nontemporal_load((const v2f*)(inpRow + kk));
        v2f b0 = *(const v2f*)&sUt[(fm +  0) * UT_STRIDE + kk];
        v2f b1 = *(const v2f*)&sUt[(fm + 16) * UT_STRIDE + kk];
        c0 = __builtin_amdgcn_wmma_f32_16x16x4_f32(false, a, false, b0,
                                                   (short)0, c0, false, false);
        c1 = __builtin_amdgcn_wmma_f32_16x16x4_f32(false, a, false, b1,
                                                   (short)0, c1, false, false);
    }

    // Spill h from WMMA C layout (VGPR g <-> row g / g+8) into padded LDS tile.
    float* hT = &sH[wave][0];
    #pragma unroll
    for (int g = 0; g < 8; ++g) {
        const int M = g + 8 * kh;
        hT[M * H_STRIDE + fm]      = c0[g];
        hT[M * H_STRIDE + fm + 16] = c1[g];
    }
    asm volatile("s_wait_dscnt 0" ::: "memory");   // same-wave LDS RAW fence

    // =====================================================================
    // Step 2: h2[m][o] = sum_i h[m][i] * W[m][i][o]   (per-sample 32x32 GEMV)
    //   lane l owns output column o = l. Each inner load is one fully
    //   coalesced 128B wave read of gen_weight (512 MB streamed, NT hint).
    // =====================================================================
    const float* gwTile = gw + (size_t)row0 * (RANK * RANK);
    #pragma unroll 2
    for (int mm = 0; mm < 16; ++mm) {
        const float* wrow = gwTile + (size_t)mm * (RANK * RANK);
        if (row0 + mm + 1 < NB)                    // prefetch next sample's 4KB
            __builtin_prefetch(wrow + RANK * RANK + lane * RANK, 0, 0);
        float acc = 0.f;
        #pragma unroll
        for (int i = 0; i < RANK; ++i) {
            float hv = hT[mm * H_STRIDE + i];      // LDS broadcast
            float wv = __builtin_nontemporal_load(wrow + i * RANK + lane);
            acc = __builtin_fmaf(hv, wv, acc);
        }
        hT[mm * H_STRIDE + lane] = acc;            // overwrite h row mm with h2 row mm
    }
    asm volatile("s_wait_dscnt 0" ::: "memory");   // same-wave LDS RAW fence

    // =====================================================================
    // Step 3: out[16x128] = h2[16x32] @ V[32x128] + bias
    //   8 K-steps x 8 N-tiles of V_WMMA_F32_16X16X4_F32 (8 indep acc chains).
    // =====================================================================
    v8f d0 = {}, d1 = {}, d2 = {}, d3 = {}, d4 = {}, d5 = {}, d6 = {}, d7 = {};
    #pragma unroll
    for (int k = 0; k < RANK / 4; ++k) {
        const int kk = 4 * k + 2 * kh;
        v2f a = *(const v2f*)&hT[fm * H_STRIDE + kk];
        v2f b;
        b = *(const v2f*)&sVt[(0 * 16 + fm) * VT_STRIDE + kk];
        d0 = __builtin_amdgcn_wmma_f32_16x16x4_f32(false, a, false, b, (short)0, d0, false, false);
        b = *(const v2f*)&sVt[(1 * 16 + fm) * VT_STRIDE + kk];
        d1 = __builtin_amdgcn_wmma_f32_16x16x4_f32(false, a, false, b, (short)0, d1, false, false);
        b = *(const v2f*)&sVt[(2 * 16 + fm) * VT_STRIDE + kk];
        d2 = __builtin_amdgcn_wmma_f32_16x16x4_f32(false, a, false, b, (short)0, d2, false, false);
        b = *(const v2f*)&sVt[(3 * 16 + fm) * VT_STRIDE + kk];
        d3 = __builtin_amdgcn_wmma_f32_16x16x4_f32(false, a, false, b, (short)0, d3, false, false);
        b = *(const v2f*)&sVt[(4 * 16 + fm) * VT_STRIDE + kk];
        d4 = __builtin_amdgcn_wmma_f32_16x16x4_f32(false, a, false, b, (short)0, d4, false, false);
        b = *(const v2f*)&sVt[(5 * 16 + fm) * VT_STRIDE + kk];
        d5 = __builtin_amdgcn_wmma_f32_16x16x4_f32(false, a, false, b, (short)0, d5, false, false);
        b = *(const v2f*)&sVt[(6 * 16 + fm) * VT_STRIDE + kk];
        d6 = __builtin_amdgcn_wmma_f32_16x16x4_f32(false, a, false, b, (short)0, d6, false, false);
        b = *(const v2f*)&sVt[(7 * 16 + fm) * VT_STRIDE + kk];
        d7 = __builtin_amdgcn_wmma_f32_16x16x4_f32(false, a, false, b, (short)0, d7, false, false);
    }

    // Epilogue: add bias, NT store (D frag: VGPR g -> row g + 8*kh, col nt*16 + fm)
    v8f dacc[8] = {d0, d1, d2, d3, d4, d5, d6, d7};
    #pragma unroll
    for (int nt = 0; nt < 8; ++nt) {
        const int n  = nt * 16 + fm;
        const float bv = sBias[n];
        #pragma unroll
        for (int g = 0; g < 8; ++g) {
            const int M = g + 8 * kh;
            __builtin_nontemporal_store(dacc[nt][g] + bv,
                                        out + (size_t)(row0 + M) * OUT_FEAT + n);
        }
    }
}

extern "C" void kernel_launch(void* const* d_in, const int* in_sizes, int n_in,
                              void* d_out, int out_size, void* d_ws, size_t ws_size,
                              hipStream_t stream) {
    (void)n_in; (void)out_size; (void)d_ws; (void)ws_size;
    const float* inp  = (const float*)d_in[0];
    const float* gw   = (const float*)d_in[1];
    const float* U    = (const float*)d_in[2];
    const float* V    = (const float*)d_in[3];
    const float* bias = (const float*)d_in[4];
    float* out = (float*)d_out;

    const int Bn   = in_sizes[0] / IN_FEAT;   // 131072
    const int grid = Bn / 128;                // 8 waves x 16 samples per block
    apg_linear_kernel<<<grid, 256, 0, stream>>>(inp, gw, U, V, bias, out);
}